// HashEmbedder3D_80530636800410
// MI455X (gfx1250) — compile-verified
//
#include <hip/hip_runtime.h>
#include <math.h>
#include <stdint.h>

// ---------------------------------------------------------------------------
// HashEmbedder3D (Instant-NGP multires hash grid), MI455X / gfx1250.
// Gather-latency-bound: one lane per (point, level), 2 points per wave32,
// 2-way point unroll per lane for 16 outstanding gathers.
// Levels 1-2 tables staged to LDS via TDM tensor_load_to_lds,
// level 0 stays in WGP$, levels 3+ hit the 192MB L2 (tables total 64MB).
// ---------------------------------------------------------------------------

#define NUM_LEVELS 16
#define HASHMAP_SZ (1u << 19)
#define PRIME0 2654435761u
#define PRIME1 805459861u
#define PRIME2 3674653429u

// Expected table sizes for LDS-staged levels (res 22 -> 22^3, res 30 -> 30^3).
#define L1_ENT 10648
#define L2_ENT 27000
#define LDS_FLOATS ((L1_ENT + L2_ENT) * 2)   // 75296 floats = 301,184 bytes

struct LevelParams {
  float    resf[NUM_LEVELS];
  int      resm1[NUM_LEVELS];
  unsigned tsize[NUM_LEVELS];
  double   dinv[NUM_LEVELS];
  int      useLds;
};

typedef float        f32x2 __attribute__((ext_vector_type(2)));
typedef unsigned int u32x4 __attribute__((ext_vector_type(4)));
typedef int          i32x8 __attribute__((ext_vector_type(8)));
typedef int          i32x4 __attribute__((ext_vector_type(4)));

#if defined(__HIP_DEVICE_COMPILE__) && __has_builtin(__builtin_amdgcn_tensor_load_to_lds)
#define HAVE_TDM 1
#else
#define HAVE_TDM 0
#endif

#if HAVE_TDM
// 1-D contiguous tile copy via Tensor Data Mover. Descriptor per ISA 08 §8.3/8.4:
// group0: count=1, lds_addr, global_addr[56:0], type=2.
// group1: data_size=8B, tensor_dim0 = tile_dim0 = nelem8, dim1=1, stride0=nelem8.
__device__ __forceinline__ void tdm_load_1d(uint32_t lds_addr, const float* gsrc,
                                            uint32_t nelem8) {
  uint64_t ga = (uint64_t)(uintptr_t)gsrc;
  u32x4 g0;
  g0[0] = 1u;                                                // count=1, user desc
  g0[1] = lds_addr;                                          // LDS byte address
  g0[2] = (uint32_t)ga;                                      // global_addr[31:0]
  g0[3] = (uint32_t)((ga >> 32) & 0x1FFFFFFu) | (2u << 30);  // addr[56:32], type=2
  i32x8 g1;
  g1[0] = (int)(3u << 16);                         // data_size=3 (8 bytes)
  g1[1] = (int)((nelem8 & 0xFFFFu) << 16);         // tensor_dim0[15:0]
  g1[2] = (int)((nelem8 >> 16) | (1u << 16));      // tensor_dim0[31:16], tensor_dim1=1
  g1[3] = (int)((nelem8 & 0xFFFFu) << 16);         // tensor_dim1 hi=0, tile_dim0
  g1[4] = 1;                                       // tile_dim1=1, tile_dim2=0
  g1[5] = (int)nelem8;                             // tensor_dim0_stride[31:0]
  g1[6] = (int)((nelem8 & 0xFFFFu) << 16);         // stride0 hi=0, stride1 lo
  g1[7] = (int)(nelem8 >> 16);                     // stride1 hi
  i32x4 z4 = {0, 0, 0, 0};                         // groups 2/3 unused (<=2D tensor)
  i32x8 z8 = {0, 0, 0, 0, 0, 0, 0, 0};
  __builtin_amdgcn_tensor_load_to_lds(g0, g1, z4, z4, z8, 0);
}
__device__ __forceinline__ void wait_tensorcnt0() {
#if __has_builtin(__builtin_amdgcn_s_wait_tensorcnt)
  __builtin_amdgcn_s_wait_tensorcnt(0);
#else
  asm volatile("s_wait_tensorcnt 0x0" ::: "memory");
#endif
}
#else
// Fallback: per-lane async global->LDS 16B copies (ASYNCcnt path).
__device__ __forceinline__ void async_copy16(uint32_t lds_byte, uint64_t gaddr) {
  asm volatile("global_load_async_to_lds_b128 %0, %1, off"
               :: "v"(lds_byte), "v"(gaddr) : "memory");
}
__device__ __forceinline__ void wait_asynccnt0() {
  asm volatile("s_wait_asynccnt 0x0" ::: "memory");
}
#endif

// Per-(point, level) corner preparation: 8 hash indices + trilinear weights.
__device__ __forceinline__ void prep_point(const float* __restrict__ x, int p,
                                           float rf, int rm1, unsigned d,
                                           int64_t dd, double dinv,
                                           uint32_t idx[8], float wc[8]) {
  const float pv[3] = { x[3 * (size_t)p + 0], x[3 * (size_t)p + 1], x[3 * (size_t)p + 2] };
  const unsigned primes[3] = { PRIME0, PRIME1, PRIME2 };
  uint64_t hh[3][2];
  float    ww[3][2];
#pragma unroll
  for (int k = 0; k < 3; ++k) {
    float xn = (pv[k] + 1.0f) * 0.5f;
    xn = fminf(fmaxf(xn, 0.0f), 1.0f);
    float xs = xn * rf;
    float f  = floorf(xs);
    int c0i  = (int)f;                 // >= 0 since xs >= 0
    int c0   = c0i < rm1 ? c0i : rm1;
    int c1i  = c0i + 1;
    int c1   = c1i < rm1 ? c1i : rm1;
    float w  = xs - f;
    hh[k][0] = (uint64_t)(uint32_t)c0 * (uint64_t)primes[k];
    hh[k][1] = (uint64_t)(uint32_t)c1 * (uint64_t)primes[k];
    ww[k][0] = 1.0f - w;
    ww[k][1] = w;
  }
#pragma unroll
  for (int c = 0; c < 8; ++c) {
    const int b0 = (c >> 2) & 1, b1 = (c >> 1) & 1, b2 = c & 1;
    uint64_t h = hh[0][b0] ^ hh[1][b1] ^ hh[2][b2];   // < 2^43, exact in double
    uint64_t q = (uint64_t)((double)h * dinv);
    int64_t  r = (int64_t)(h - q * (uint64_t)d);
    r = (r < 0)   ? r + dd : r;
    r = (r >= dd) ? r - dd : r;
    idx[c] = (uint32_t)r;
    wc[c]  = ww[0][b0] * ww[1][b1] * ww[2][b2];
  }
}

__global__ __launch_bounds__(1024)
void hash_encode_kernel(const float* __restrict__ x,
                        const float* __restrict__ tables,
                        float* __restrict__ out,
                        int N, LevelParams lp) {
  __shared__ __attribute__((aligned(16))) float s_tbl[LDS_FLOATS];

  if (lp.useLds) {
#if HAVE_TDM
    if (threadIdx.x == 0) {
      tdm_load_1d((uint32_t)(uintptr_t)&s_tbl[0],
                  tables + (size_t)1 * HASHMAP_SZ * 2, L1_ENT);
      tdm_load_1d((uint32_t)(uintptr_t)&s_tbl[L1_ENT * 2],
                  tables + (size_t)2 * HASHMAP_SZ * 2, L2_ENT);
      wait_tensorcnt0();
    }
#else
    {
      const uint64_t src1 = (uint64_t)(uintptr_t)(tables + (size_t)1 * HASHMAP_SZ * 2);
      const uint64_t src2 = (uint64_t)(uintptr_t)(tables + (size_t)2 * HASHMAP_SZ * 2);
      const uint32_t lds1 = (uint32_t)(uintptr_t)&s_tbl[0];
      const uint32_t lds2 = (uint32_t)(uintptr_t)&s_tbl[L1_ENT * 2];
      for (uint32_t i = threadIdx.x; i < (L1_ENT * 8u) / 16u; i += blockDim.x)
        async_copy16(lds1 + i * 16u, src1 + (uint64_t)i * 16u);
      for (uint32_t i = threadIdx.x; i < (L2_ENT * 8u) / 16u; i += blockDim.x)
        async_copy16(lds2 + i * 16u, src2 + (uint64_t)i * 16u);
      wait_asynccnt0();
    }
#endif
    __syncthreads();
  }

  const int level   = (int)(threadIdx.x & 15);
  const float rf    = lp.resf[level];
  const int rm1     = lp.resm1[level];
  const unsigned d  = lp.tsize[level];
  const int64_t dd  = (int64_t)d;
  const double dinv = lp.dinv[level];
  const f32x2* __restrict__ gtab = (const f32x2*)tables + (size_t)level * HASHMAP_SZ;
  const float* ltab = (level == 1) ? &s_tbl[0] : &s_tbl[L1_ENT * 2];
  const bool fromLds = lp.useLds && (level == 1 || level == 2);

  int p = (int)((blockIdx.x * blockDim.x + threadIdx.x) >> 4);
  const int pstep = (int)((gridDim.x * blockDim.x) >> 4);
  f32x2* __restrict__ o = (f32x2*)out;

  // 2-way point unroll: 16 outstanding gathers per lane.
  for (; p < N; p += 2 * pstep) {
    const int  pA   = p;
    const int  pBr  = p + pstep;
    const bool hasB = pBr < N;
    const int  pB   = hasB ? pBr : pA;    // safe address; store predicated below

    uint32_t idxA[8], idxB[8];
    float    wcA[8],  wcB[8];
    prep_point(x, pA, rf, rm1, d, dd, dinv, idxA, wcA);
    prep_point(x, pB, rf, rm1, d, dd, dinv, idxB, wcB);

    f32x2 accA = { 0.0f, 0.0f };
    f32x2 accB = { 0.0f, 0.0f };
    if (fromLds) {
      f32x2 gA[8], gB[8];
#pragma unroll
      for (int c = 0; c < 8; ++c) gA[c] = *(const f32x2*)(ltab + 2 * (size_t)idxA[c]);
#pragma unroll
      for (int c = 0; c < 8; ++c) gB[c] = *(const f32x2*)(ltab + 2 * (size_t)idxB[c]);
#pragma unroll
      for (int c = 0; c < 8; ++c) {
        accA.x += wcA[c] * gA[c].x; accA.y += wcA[c] * gA[c].y;
        accB.x += wcB[c] * gB[c].x; accB.y += wcB[c] * gB[c].y;
      }
    } else {
      f32x2 gA[8], gB[8];
#pragma unroll
      for (int c = 0; c < 8; ++c) gA[c] = gtab[idxA[c]];
#pragma unroll
      for (int c = 0; c < 8; ++c) gB[c] = gtab[idxB[c]];
#pragma unroll
      for (int c = 0; c < 8; ++c) {
        accA.x += wcA[c] * gA[c].x; accA.y += wcA[c] * gA[c].y;
        accB.x += wcB[c] * gB[c].x; accB.y += wcB[c] * gB[c].y;
      }
    }
    __builtin_nontemporal_store(accA, &o[(size_t)pA * NUM_LEVELS + level]);
    if (hasB)
      __builtin_nontemporal_store(accB, &o[(size_t)pBr * NUM_LEVELS + level]);
  }
}

extern "C" void kernel_launch(void* const* d_in, const int* in_sizes, int n_in,
                              void* d_out, int out_size, void* d_ws, size_t ws_size,
                              hipStream_t stream) {
  const float* x      = (const float*)d_in[0];
  const float* tables = (const float*)d_in[1];
  float* out          = (float*)d_out;
  const int N = in_sizes[0] / 3;

  LevelParams lp;
  const double b = exp((log(2048.0) - log(16.0)) / 15.0);
  for (int i = 0; i < NUM_LEVELS; ++i) {
    int res = (int)floor(16.0 * pow(b, (double)i));   // same double math as numpy
    long long t3 = (long long)res * res * res;
    unsigned ts = (t3 < (long long)HASHMAP_SZ) ? (unsigned)t3 : HASHMAP_SZ;
    lp.resf[i]  = (float)res;
    lp.resm1[i] = res - 1;
    lp.tsize[i] = ts;
    lp.dinv[i]  = 1.0 / (double)ts;
  }
  lp.useLds = (lp.tsize[1] == L1_ENT && lp.tsize[2] == L2_ENT) ? 1 : 0;

  const int threads = 1024;            // 32 waves -> 8 waves/SIMD at 1 WG/WGP (301KB LDS)
  const int blocks  = 512;             // grid-stride over 2M points
  hipLaunchKernelGGL(hash_encode_kernel, dim3(blocks), dim3(threads), 0, stream,
                     x, tables, out, N, lp);
}